// DocumentModel_25297357373868
// MI455X (gfx1250) — compile-verified
//
#include <hip/hip_runtime.h>

#define SEQ 1024
#define TPB 1024

typedef __attribute__((ext_vector_type(4))) float        v4f;
typedef __attribute__((ext_vector_type(4))) unsigned int v4u;

// AMDGCN address-space-qualified element types for the CDNA5 async builtins.
// (clang's diagnostic shows the builtin takes AS-qualified int*:
//  "__device__ int *" == addrspace(1) int*, "__shared__ int*" == addrspace(3) int*)
typedef __attribute__((address_space(1))) int gint_t;   // global
typedef __attribute__((address_space(3))) int lint_t;   // LDS

__global__ __launch_bounds__(TPB) void tfidf_kernel(const int* __restrict__ x,
                                                    const float* __restrict__ idf,
                                                    float* __restrict__ out,
                                                    int V, int histPad) {
    extern __shared__ char smem[];
    // LDS layout (all 16B aligned): hist[histPad] u32 | toks[SEQ] i32 | partials[32] f32 | inv f32
    unsigned int* hist     = (unsigned int*)smem;
    int*          toks     = (int*)  (smem + (size_t)histPad * 4);
    float*        partials = (float*)(smem + (size_t)histPad * 4 + SEQ * 4);
    float*        invp     = (float*)(smem + (size_t)histPad * 4 + SEQ * 4 + 128);

    const int b   = blockIdx.x;
    const int tid = threadIdx.x;
    const int* xrow = x + (size_t)b * SEQ;

    // --- Phase A: stage this row's tokens into LDS via the CDNA5 async
    // global->LDS path (tracked by ASYNCcnt), overlapping the copy with the
    // LDS histogram zeroing below.
#if __has_builtin(__builtin_amdgcn_global_load_async_to_lds_b32)
    __builtin_amdgcn_global_load_async_to_lds_b32(
        (gint_t*)(xrow + tid), (lint_t*)(toks + tid), 0, 0);
#else
    toks[tid] = xrow[tid];
#endif

    // Zero the histogram with vector LDS stores (ds_store_b128).
    const int nzero4 = histPad >> 2;
    v4u z; z.x = 0u; z.y = 0u; z.z = 0u; z.w = 0u;
    for (int i = tid; i < nzero4; i += TPB) ((v4u*)hist)[i] = z;

#if __has_builtin(__builtin_amdgcn_s_wait_asynccnt)
    __builtin_amdgcn_s_wait_asynccnt(0);
#endif
    __syncthreads();

    // --- Phase B: LDS scatter-add histogram + deterministic row-sum of idf.
    const int tok = toks[tid];
    atomicAdd(&hist[tok], 1u);            // ds_add_u32 (no return)
    float val = idf[tok];                 // gather, L2-hot

    // wave32 shuffle reduction
    #pragma unroll
    for (int off = 16; off > 0; off >>= 1) val += __shfl_down(val, off, 32);
    const int lane = tid & 31;
    const int wave = tid >> 5;
    if (lane == 0) partials[wave] = val;
    __syncthreads();
    if (tid < 32) {
        float p = partials[tid];          // exactly 32 waves in this block
        #pragma unroll
        for (int off = 16; off > 0; off >>= 1) p += __shfl_down(p, off, 32);
        if (tid == 0) *invp = 1.0f / p;
    }
    __syncthreads();
    const float inv = *invp;

    // --- Phase C: stream the dense row out = count * idf * inv.
    // Row base element offset b*V is not a multiple of 4 (V odd), so peel a
    // head of `lead` scalars to 16B-align the float4 main body.
    float* orow = out + (size_t)b * V;
    const int lead = (int)((4u - (unsigned)(((size_t)b * (size_t)V) & 3u)) & 3u);

    if (tid < lead)
        orow[tid] = (float)hist[tid] * idf[tid] * inv;

    const int nvec = (V - lead) >> 2;
    for (int k = tid; k < nvec; k += TPB) {
        const int v0 = lead + 4 * k;
        const unsigned c0 = hist[v0 + 0];
        const unsigned c1 = hist[v0 + 1];
        const unsigned c2 = hist[v0 + 2];
        const unsigned c3 = hist[v0 + 3];
        v4f r;
        r.x = (float)c0 * idf[v0 + 0] * inv;
        r.y = (float)c1 * idf[v0 + 1] * inv;
        r.z = (float)c2 * idf[v0 + 2] * inv;
        r.w = (float)c3 * idf[v0 + 3] * inv;
        *(v4f*)(orow + v0) = r;           // global_store_b128, 16B aligned
    }

    for (int v = lead + 4 * nvec + tid; v < V; v += TPB)
        orow[v] = (float)hist[v] * idf[v] * inv;
}

extern "C" void kernel_launch(void* const* d_in, const int* in_sizes, int n_in,
                              void* d_out, int out_size, void* d_ws, size_t ws_size,
                              hipStream_t stream) {
    const int*   x   = (const int*)d_in[0];    // (B, SEQ) token ids
    const float* idf = (const float*)d_in[1];  // (V,)
    float*       out = (float*)d_out;          // (B, V)

    const int V = in_sizes[1];
    const int B = in_sizes[0] / SEQ;
    const int histPad = (V + 3) & ~3;

    const size_t shmem = (size_t)histPad * 4 + (size_t)SEQ * 4 + 128 + 16;

    // ~205 KB dynamic LDS: legal on CDNA5 (320 KB/WGP) but above the default
    // cap, so raise the attribute. Capture-safe (no stream work enqueued).
    (void)hipFuncSetAttribute(reinterpret_cast<const void*>(tfidf_kernel),
                              hipFuncAttributeMaxDynamicSharedMemorySize, (int)shmem);

    tfidf_kernel<<<B, TPB, shmem, stream>>>(x, idf, out, V, histPad);
}